// neural_rendering_raster_and_texture_61993557950630
// MI455X (gfx1250) — compile-verified
//
#include <hip/hip_runtime.h>
#include <math.h>

#define IMG_H 128
#define IMG_W 128
#define BATCH 4
#define NVERT 4096
#define NFACE 8192
#define TEXH 2048
#define TEXW 2048
#define EPSA 1e-9f
#define WEPS 1e-6f

#define CHUNK_FACES 64                      // faces staged per TDM transfer
#define CHUNK_DW    (CHUNK_FACES * 16)      // 1024 dwords = 4KB
#define NCHUNK      (NFACE / CHUNK_FACES)   // 128

typedef __attribute__((ext_vector_type(2))) float v2f;
typedef __attribute__((ext_vector_type(8))) float v8f;
typedef __attribute__((ext_vector_type(4))) unsigned int v4u;
typedef __attribute__((ext_vector_type(8))) int v8i;
typedef __attribute__((ext_vector_type(4))) int v4i;

// ---------------------------------------------------------------------------
// Stage 1: vertex transform -> screen coords (x, y, z_ndc, w) per vertex
// ---------------------------------------------------------------------------
__global__ void k_transform(const float* __restrict__ verts,
                            const float* __restrict__ poses,
                            float* __restrict__ svert) {
  int tid = blockIdx.x * blockDim.x + threadIdx.x;
  if (tid >= BATCH * NVERT) return;
  int b = tid / NVERT, n = tid % NVERT;
  const float* P = poses + b * 16;
  const float* V = verts + (size_t)(b * NVERT + n) * 3;
  float x = V[0], y = V[1], z = V[2];
  // cam_pose: rows of pose[:3,:3] premultiplied by G=diag(1,-1,-1); t_y,t_z negated
  float c0 =  (P[0] * x + P[1] * y + P[2]  * z + P[3]);
  float c1 = -(P[4] * x + P[5] * y + P[6]  * z + P[7]);
  float c2 = -(P[8] * x + P[9] * y + P[10] * z + P[11]);
  float c3 =  (P[12] * x + P[13] * y + P[14] * z + P[15]);
  const float RIGHT = (IMG_W - 1.0f) / 2.0f * 0.1f / 140.0f;
  const float TOPV  = RIGHT * ((float)IMG_H / (float)IMG_W);
  const float P00 = 0.1f / RIGHT;
  const float P11 = 0.1f / TOPV;
  const float P22 = -(10.0f + 0.1f) / (10.0f - 0.1f);
  const float P32 = -2.0f * 10.0f * 0.1f / (10.0f - 0.1f);
  float cx = c0 * P00;
  float cy = c1 * P11;
  float cz = c2 * P22 + c3 * P32;
  float cw = -c2;
  float4 s;
  s.x = (cx / cw * 0.5f + 0.5f) * (float)IMG_W;
  s.y = (0.5f - cy / cw * 0.5f) * (float)IMG_H;
  s.z = cz / cw;
  s.w = cw;
  ((float4*)svert)[b * NVERT + n] = s;
}

// ---------------------------------------------------------------------------
// Stage 2a: per-(batch,face) affine edge coefficients
// record = [a0,b0,c0, a1,b1,c1, a2,b2,c2, az,bz,cz, ok, 0,0,0]  (64 bytes)
// e_i(px,py) = a_i*px + b_i*py + c_i ; ez = e0*z0+e1*z1+e2*z2
// ---------------------------------------------------------------------------
__global__ void k_facecoef(const int* __restrict__ faces,
                           const float* __restrict__ svert,
                           float* __restrict__ frec) {
  int tid = blockIdx.x * blockDim.x + threadIdx.x;
  if (tid >= BATCH * NFACE) return;
  int b = tid / NFACE, f = tid % NFACE;
  int i0 = faces[f * 3 + 0], i1 = faces[f * 3 + 1], i2 = faces[f * 3 + 2];
  const float4* sv = ((const float4*)svert) + b * NVERT;
  float4 v0 = sv[i0], v1 = sv[i1], v2 = sv[i2];
  float a0 = -(v2.y - v1.y), b0 = (v2.x - v1.x);
  float c0 = (v2.y - v1.y) * v1.x - (v2.x - v1.x) * v1.y;
  float a1 = -(v0.y - v2.y), b1 = (v0.x - v2.x);
  float c1 = (v0.y - v2.y) * v2.x - (v0.x - v2.x) * v2.y;
  float a2 = -(v1.y - v0.y), b2 = (v1.x - v0.x);
  float c2 = (v1.y - v0.y) * v0.x - (v1.x - v0.x) * v0.y;
  float az = a0 * v0.z + a1 * v1.z + a2 * v2.z;
  float bz = b0 * v0.z + b1 * v1.z + b2 * v2.z;
  float cz = c0 * v0.z + c1 * v1.z + c2 * v2.z;
  float ok = (v0.w > WEPS && v1.w > WEPS && v2.w > WEPS) ? 1.0f : 0.0f;
  float4* R = ((float4*)frec) + (size_t)tid * 4;
  R[0] = make_float4(a0, b0, c0, a1);
  R[1] = make_float4(b1, c1, a2, b2);
  R[2] = make_float4(c2, az, bz, cz);
  R[3] = make_float4(ok, 0.0f, 0.0f, 0.0f);
}

// ---------------------------------------------------------------------------
// TDM helper: issue a 4KB (1024-dword) contiguous global->LDS tensor copy.
// D# per CDNA5 ISA §8:
//   group0: count=1 | lds_addr | global_addr[56:0] | type=2
//   group1: data_size=4B | tensor_dim0=1024 | tile_dim0=1024 (1-D tile)
// ---------------------------------------------------------------------------
__device__ __forceinline__ void tdm_load_4k(unsigned lds_addr, unsigned long long gaddr) {
  v4u g0;
  g0.x = 1u;                                   // count=1 (valid), user mode
  g0.y = lds_addr;                             // lds_addr (bytes)
  g0.z = (unsigned)gaddr;                      // global_addr[31:0]
  g0.w = (unsigned)((gaddr >> 32) & 0x1FFFFFFu) | (2u << 30);  // ga[56:32] | type=2
  v8i g1;
  g1[0] = (int)(2u << 16);                     // data_size=2 (4 bytes); no multicast
  g1[1] = (int)(1024u << 16);                  // tensor_dim0[15:0]=1024
  g1[2] = (int)(1u << 16);                     // tensor_dim0 hi=0 | tensor_dim1=1
  g1[3] = (int)(1024u << 16);                  // tensor_dim1 hi=0 | tile_dim0=1024
  g1[4] = 0;                                   // tile_dim1=0 (unused), tile_dim2=0
  g1[5] = 1024;                                // tensor_dim0_stride
  g1[6] = 0;
  g1[7] = 0;
  v4i zz; zz[0] = 0; zz[1] = 0; zz[2] = 0; zz[3] = 0;
#if __clang_major__ >= 23
  v8i z8; z8[0]=0; z8[1]=0; z8[2]=0; z8[3]=0; z8[4]=0; z8[5]=0; z8[6]=0; z8[7]=0;
  __builtin_amdgcn_tensor_load_to_lds(g0, g1, zz, zz, z8, 0);
#else
  __builtin_amdgcn_tensor_load_to_lds(g0, g1, zz, zz, 0);
#endif
}

// ---------------------------------------------------------------------------
// Stage 2b: WMMA rasterizer.
// Block = 256 threads = 8 waves = one full image row of one batch (8 x 16px
// tiles) -> all waves consume the same face stream, staged through LDS by the
// Tensor Data Mover (double-buffered 2x4KB, s_wait_tensorcnt + barriers).
// Per 16-face sub-block: 4x V_WMMA_F32_16X16X4_F32 evaluate e0,e1,e2,ez for
// 16 pixels x 16 faces. C element (vgpr r, lane l): pixel = r + 8*(l>=16),
// face = l&15. Each lane accumulates per-pixel (min z, face) over its face
// residue class; one 16-lane butterfly reduction at the very end.
// ---------------------------------------------------------------------------
__global__ void __launch_bounds__(256) k_raster(const float* __restrict__ frec,
                                                int* __restrict__ face_id) {
  __shared__ __align__(16) float ldsrec[2][CHUNK_DW];

  int wave = threadIdx.x >> 5;           // 0..7 -> tile x
  int lane = threadIdx.x & 31;
  int row = blockIdx.x % IMG_H;
  int b   = blockIdx.x / IMG_H;
  int tx  = wave;

  bool lo = lane < 16;
  int  nf = lane & 15;
  float px = (float)(tx * 16 + nf) + 0.5f;
  float py = (float)row + 0.5f;
  // A (16x4, pixels x [px,py,1,0]): lanes 0-15 hold K0,K1 ; lanes 16-31 K2,K3
  v2f Av; Av.x = lo ? px : 1.0f; Av.y = lo ? py : 0.0f;

  float bestz[8]; int bestf[8];
#pragma unroll
  for (int r = 0; r < 8; ++r) { bestz[r] = __builtin_inff(); bestf[r] = -1; }

  const unsigned long long gbase =
      (unsigned long long)(uintptr_t)(frec + (size_t)b * NFACE * 16);
  bool issuer = (threadIdx.x < 32);      // wave 0 drives the TDM

  if (issuer) {
    tdm_load_4k((unsigned)(uintptr_t)&ldsrec[0][0], gbase);
  }

  for (int c = 0; c < NCHUNK; ++c) {
    if (issuer) {
      if (c + 1 < NCHUNK) {
        tdm_load_4k((unsigned)(uintptr_t)&ldsrec[(c + 1) & 1][0],
                    gbase + (unsigned long long)(c + 1) * (CHUNK_DW * 4));
        __builtin_amdgcn_s_wait_tensorcnt(1);  // in-order: chunk c is done
      } else {
        __builtin_amdgcn_s_wait_tensorcnt(0);
      }
    }
    __syncthreads();                     // chunk c visible to all waves

    const float* buf = &ldsrec[c & 1][0];
#pragma unroll
    for (int sub = 0; sub < 4; ++sub) {
      const float4* rc = (const float4*)(buf + (size_t)(sub * 16 + nf) * 16);
      float4 r0 = rc[0], r1 = rc[1], r2 = rc[2], r3 = rc[3];
      // B (4x16): lanes 0-15 hold K0=a,K1=b ; lanes 16-31 hold K2=c,K3=0
      v2f B0, B1, B2, B3;
      B0.x = lo ? r0.x : r0.z;  B0.y = lo ? r0.y : 0.0f;  // e0
      B1.x = lo ? r0.w : r1.y;  B1.y = lo ? r1.x : 0.0f;  // e1
      B2.x = lo ? r1.z : r2.x;  B2.y = lo ? r1.w : 0.0f;  // e2
      B3.x = lo ? r2.y : r2.w;  B3.y = lo ? r2.z : 0.0f;  // ez
      v8f cz8 = {};
      v8f ce0 = __builtin_amdgcn_wmma_f32_16x16x4_f32(false, Av, false, B0, (short)0, cz8, false, false);
      v8f ce1 = __builtin_amdgcn_wmma_f32_16x16x4_f32(false, Av, false, B1, (short)0, cz8, false, false);
      v8f ce2 = __builtin_amdgcn_wmma_f32_16x16x4_f32(false, Av, false, B2, (short)0, cz8, false, false);
      v8f cez = __builtin_amdgcn_wmma_f32_16x16x4_f32(false, Av, false, B3, (short)0, cz8, false, false);
      bool okw = r3.x > 0.5f;
      int idx = c * CHUNK_FACES + sub * 16 + nf;
#pragma unroll
      for (int r = 0; r < 8; ++r) {
        float e0 = ce0[r], e1 = ce1[r], e2 = ce2[r], ez = cez[r];
        float area = e0 + e1 + e2;
        float s = (area > 0.0f) ? 1.0f : ((area < 0.0f) ? -1.0f : 0.0f);
        bool okA = fabsf(area) > EPSA;
        float zp = ez / (okA ? area : 1.0f);
        bool inside = okw && okA &&
                      (e0 * s >= 0.0f) && (e1 * s >= 0.0f) && (e2 * s >= 0.0f) &&
                      (zp >= -1.0f) && (zp <= 1.0f);
        float zc = inside ? zp : __builtin_inff();
        // per-lane accumulate: indices increase, so strict < keeps earliest tie
        if (zc < bestz[r]) { bestz[r] = zc; bestf[r] = idx; }
      }
    }
    __syncthreads();                     // done reading before buffer reuse
  }

  // Final min-z reduction over the 16 faces residues (tie -> lowest index),
  // within each 16-lane half.
#pragma unroll
  for (int r = 0; r < 8; ++r) {
    float zc = bestz[r]; int idx = bestf[r];
#pragma unroll
    for (int off = 1; off < 16; off <<= 1) {
      float zo = __shfl_xor(zc, off, 32);
      int   io = __shfl_xor(idx, off, 32);
      if (zo < zc || (zo == zc && io < idx)) { zc = zo; idx = io; }
    }
    bestz[r] = zc; bestf[r] = idx;
  }

  int base = (b * IMG_H + row) * IMG_W + tx * 16;
  if (lane == 0) {
#pragma unroll
    for (int r = 0; r < 8; ++r) face_id[base + r] = bestf[r];
  }
  if (lane == 16) {
#pragma unroll
    for (int r = 0; r < 8; ++r) face_id[base + 8 + r] = bestf[r];
  }
}

// ---------------------------------------------------------------------------
// Stage 3+4: barycentric shade + bilinear texture sample, one thread/pixel
// ---------------------------------------------------------------------------
__global__ void k_shade(const int* __restrict__ face_id,
                        const int* __restrict__ faces,
                        const float* __restrict__ svert,
                        const float* __restrict__ uvmap,
                        const float* __restrict__ tex,
                        float* __restrict__ out) {
  int tid = blockIdx.x * blockDim.x + threadIdx.x;
  if (tid >= BATCH * IMG_H * IMG_W) return;
  int xp = tid % IMG_W;
  int yp = (tid / IMG_W) % IMG_H;
  int b  = tid / (IMG_W * IMG_H);
  float* o = out + (size_t)tid * 3;
  int fid = face_id[tid];
  if (fid < 0) { o[0] = 0.0f; o[1] = 0.0f; o[2] = 0.0f; return; }
  int i0 = faces[fid * 3], i1 = faces[fid * 3 + 1], i2 = faces[fid * 3 + 2];
  const float4* sv = ((const float4*)svert) + b * NVERT;
  float4 v0 = sv[i0], v1 = sv[i1], v2 = sv[i2];
  float px = (float)xp + 0.5f, py = (float)yp + 0.5f;
  float e0 = (v2.x - v1.x) * (py - v1.y) - (v2.y - v1.y) * (px - v1.x);
  float e1 = (v0.x - v2.x) * (py - v2.y) - (v0.y - v2.y) * (px - v2.x);
  float e2 = (v1.x - v0.x) * (py - v0.y) - (v1.y - v0.y) * (px - v0.x);
  float bw0 = e0 / v0.w, bw1 = e1 / v1.w, bw2 = e2 / v2.w;
  float denom = bw0 + bw1 + bw2;
  if (!(fabsf(denom) > EPSA)) denom = 1.0f;
  float pc0 = bw0 / denom, pc1 = bw1 / denom, pc2 = bw2 / denom;
  float maskr = pc0 + pc1 + pc2;
  const float* U = uvmap + (size_t)b * NVERT * 2;
  float u = pc0 * U[i0 * 2]     + pc1 * U[i1 * 2]     + pc2 * U[i2 * 2];
  float v = pc0 * U[i0 * 2 + 1] + pc1 * U[i1 * 2 + 1] + pc2 * U[i2 * 2 + 1];
  // bilinear texture sample: idx = clip(uv[::-1],0,1) * [TH,TW]
  float ty  = fminf(fmaxf(v, 0.0f), 1.0f) * (float)TEXH;
  float txc = fminf(fmaxf(u, 0.0f), 1.0f) * (float)TEXW;
  float fly = floorf(ty), flx = floorf(txc);
  float fy = ty - fly, fx = txc - flx;
  int iy = (int)fly, ix = (int)flx;
  int iy0 = min(max(iy, 0), TEXH - 1), iy1 = min(max(iy + 1, 0), TEXH - 1);
  int ix0 = min(max(ix, 0), TEXW - 1), ix1 = min(max(ix + 1, 0), TEXW - 1);
  const float* T = tex + (size_t)b * TEXH * TEXW * 3;
  const float* tl = T + ((size_t)iy0 * TEXW + ix0) * 3;
  const float* tr = T + ((size_t)iy0 * TEXW + ix1) * 3;
  const float* bl = T + ((size_t)iy1 * TEXW + ix0) * 3;
  const float* br = T + ((size_t)iy1 * TEXW + ix1) * 3;
  float w00 = (1.0f - fx) * (1.0f - fy), w10 = fx * (1.0f - fy);
  float w01 = (1.0f - fx) * fy,          w11 = fx * fy;
#pragma unroll
  for (int c = 0; c < 3; ++c)
    o[c] = (tl[c] * w00 + tr[c] * w10 + bl[c] * w01 + br[c] * w11) * maskr;
}

// ---------------------------------------------------------------------------
extern "C" void kernel_launch(void* const* d_in, const int* in_sizes, int n_in,
                              void* d_out, int out_size, void* d_ws, size_t ws_size,
                              hipStream_t stream) {
  const float* vertices = (const float*)d_in[0];
  const float* uv_map   = (const float*)d_in[1];
  const int*   faces    = (const int*)d_in[2];
  const float* texture  = (const float*)d_in[3];
  const float* poses    = (const float*)d_in[4];
  float* out = (float*)d_out;

  // Workspace layout (floats): svert[B*N*4] | frec[B*F*16] | face_id[B*H*W]
  float* svert = (float*)d_ws;
  float* frec  = svert + (size_t)BATCH * NVERT * 4;
  int*   fidb  = (int*)(frec + (size_t)BATCH * NFACE * 16);

  int nTrans = BATCH * NVERT;
  k_transform<<<(nTrans + 255) / 256, 256, 0, stream>>>(vertices, poses, svert);

  int nFace = BATCH * NFACE;
  k_facecoef<<<(nFace + 255) / 256, 256, 0, stream>>>(faces, svert, frec);

  // One block per (batch, image row): 8 waves = 8 x 16-pixel tiles
  k_raster<<<BATCH * IMG_H, 256, 0, stream>>>(frec, fidb);

  int nPix = BATCH * IMG_H * IMG_W;
  k_shade<<<(nPix + 255) / 256, 256, 0, stream>>>(fidb, faces, svert, uv_map, texture, out);
}